// LearnedRotation_1305670058162
// MI455X (gfx1250) — compile-verified
//
#include <hip/hip_runtime.h>
#include <math.h>

// ---------------------------------------------------------------------------
// LearnedRotation for MI455X (gfx1250, wave32).
//
// Memory-bound: ~0.8 GB of HBM traffic -> ~35us at 23.3 TB/s is the floor.
// Strategy: lane == channel (C=32 == wave32) so every pixel access is one
// fully-coalesced 128B transaction. The only real matmul (mean_feat @ Wm)
// is done with V_WMMA_F32_16X16X4_F32 on the matrix pipe.
// ---------------------------------------------------------------------------

typedef __attribute__((ext_vector_type(2))) float v2f;
typedef __attribute__((ext_vector_type(8))) float v8f;

#define B_  32
#define H_  256
#define W_  256
#define C_  32
#define HW_ (H_*W_)
#define CHUNKS 32
#define PIX_PER_CHUNK (HW_/CHUNKS)      // 2048 pixels per (batch, chunk) block
#define PIX_PER_WAVE  (PIX_PER_CHUNK/8) // 256 pixels per wave

// ---------------------------------------------------------------------------
// Kernel 1: per-(batch, chunk) partial channel sums. Deterministic (no
// atomics): partial[chunk][b*32+c] written exactly once.
// ---------------------------------------------------------------------------
__global__ void LR_mean_partial(const float* __restrict__ x,
                                float* __restrict__ partial) {
  const int b     = blockIdx.y;
  const int chunk = blockIdx.x;
  const int lane  = threadIdx.x & 31;   // channel
  const int wave  = threadIdx.x >> 5;   // 0..7
  const size_t base = (size_t)b * HW_ * C_;
  const int p0 = chunk * PIX_PER_CHUNK + wave * PIX_PER_WAVE;
  const float* px = x + base + (size_t)p0 * C_ + lane;

  float acc = 0.f;
#pragma unroll 8
  for (int i = 0; i < PIX_PER_WAVE; ++i)
    acc += px[(size_t)i * C_];          // 128B coalesced per wave-iter

  __shared__ float red[256];
  red[threadIdx.x] = acc;
  __syncthreads();
  if (wave == 0) {
    float s = red[lane];
#pragma unroll
    for (int w = 1; w < 8; ++w) s += red[w * 32 + lane];
    partial[(size_t)chunk * (B_ * C_) + b * C_ + lane] = s;
  }
}

// ---------------------------------------------------------------------------
// Kernel 2: deterministic fixed-order reduction of the 32 chunk partials.
// ---------------------------------------------------------------------------
__global__ void LR_reduce_chunks(const float* __restrict__ partial,
                                 float* __restrict__ sums) {
  const int idx = blockIdx.x * blockDim.x + threadIdx.x;  // 0..1023
  float s = 0.f;
#pragma unroll 8
  for (int c = 0; c < CHUNKS; ++c)
    s += partial[(size_t)c * (B_ * C_) + idx];
  sums[idx] = s;
}

// ---------------------------------------------------------------------------
// Kernel 3: one wave. mean = sums/HW; y = mean_feat @ Wm via
// V_WMMA_F32_16X16X4_F32 (2 M-tiles x 8 K-steps); angle = tanh(y+b)*pi;
// emit per-batch {cos, sin, x_off, y_off}.
//
// A layout (ISA 16x4 f32): lanes 0-15: M=lane, VGPR0/1 = K=k0+0/k0+1;
//                          lanes 16-31: M=lane-16, VGPR0/1 = K=k0+2/k0+3.
// B laid out mirrored (N across lanes); only column N=0 (== Wm) nonzero.
// D (16x16 f32): column N=0 lives in lanes 0 (M=0..7) and 16 (M=8..15).
// ---------------------------------------------------------------------------
__global__ void LR_angle(const float* __restrict__ Wm,
                         const float* __restrict__ bias,
                         const float* __restrict__ sums,
                         float* __restrict__ params) {
  __shared__ float feats[B_ * C_];
  __shared__ float yv[B_];
  const int lane = threadIdx.x;          // 0..31, single wave
  const float invHW = 1.0f / (float)HW_;
  for (int i = lane; i < B_ * C_; i += 32) feats[i] = sums[i] * invHW;
  __syncthreads();

  const int m     = lane & 15;
  const int khalf = (lane >> 4) << 1;    // 0 for lanes 0-15, 2 for 16-31

  v8f d0 = {0.f, 0.f, 0.f, 0.f, 0.f, 0.f, 0.f, 0.f};
  v8f d1 = {0.f, 0.f, 0.f, 0.f, 0.f, 0.f, 0.f, 0.f};
#pragma unroll
  for (int k0 = 0; k0 < C_; k0 += 4) {
    const int kk = k0 + khalf;
    v2f a0; a0.x = feats[m * C_ + kk];        a0.y = feats[m * C_ + kk + 1];
    v2f a1; a1.x = feats[(m + 16) * C_ + kk]; a1.y = feats[(m + 16) * C_ + kk + 1];
    v2f bf; bf.x = 0.f; bf.y = 0.f;
    if (m == 0) { bf.x = Wm[kk]; bf.y = Wm[kk + 1]; }   // column N=0 only
    d0 = __builtin_amdgcn_wmma_f32_16x16x4_f32(false, a0, false, bf,
                                               (short)0, d0, false, false);
    d1 = __builtin_amdgcn_wmma_f32_16x16x4_f32(false, a1, false, bf,
                                               (short)0, d1, false, false);
  }

  if (m == 0) {
    const int mb = (lane >> 4) * 8;      // lane0 -> M=0..7, lane16 -> M=8..15
#pragma unroll
    for (int j = 0; j < 8; ++j) {
      yv[mb + j]      = d0[j];           // batches 0..15
      yv[16 + mb + j] = d1[j];           // batches 16..31
    }
  }
  __syncthreads();

  const float PI = 3.14159265358979323846f;
  const float ang = tanhf(yv[lane] + bias[0]) * PI;   // |ang| <= pi, clip is no-op
  const float c = cosf(ang), s = sinf(ang);
  const float w1 = (float)(W_ - 1), h1 = (float)(H_ - 1);
  params[lane * 4 + 0] = c;
  params[lane * 4 + 1] = s;
  params[lane * 4 + 2] = (w1 - (c * w1 - s * h1)) * 0.5f;  // x_off
  params[lane * 4 + 3] = (h1 - (s * w1 + c * h1)) * 0.5f;  // y_off
}

// ---------------------------------------------------------------------------
// Kernel 4: bilinear rotation gather. grid=(B, H), 8 waves/block; each wave
// covers 32 consecutive output x's; lane == channel so each of the 4 taps
// and the store is one 128B coalesced transaction. Validity is wave-uniform.
// ---------------------------------------------------------------------------
__global__ void LR_rotate(const float* __restrict__ x,
                          const float* __restrict__ params,
                          float* __restrict__ out) {
  const int b    = blockIdx.x;
  const int y    = blockIdx.y;
  const int lane = threadIdx.x & 31;
  const int wave = threadIdx.x >> 5;

  const float cs   = params[b * 4 + 0];
  const float sn   = params[b * 4 + 1];
  const float xoff = params[b * 4 + 2];
  const float yoff = params[b * 4 + 3];

  const float* img  = x   + (size_t)b * HW_ * C_;
  float*       orow = out + (size_t)b * HW_ * C_ + (size_t)y * W_ * C_;
  const float fy = (float)y;

  const int x_begin = wave * (W_ / 8);
  for (int xi = x_begin; xi < x_begin + W_ / 8; ++xi) {
    const float fx   = (float)xi;
    const float in_x = cs * fx - sn * fy + xoff;
    const float in_y = sn * fx + cs * fy + yoff;
    const float x0f = floorf(in_x), y0f = floorf(in_y);
    const float wx = in_x - x0f,  wy = in_y - y0f;
    const int x0 = (int)x0f, y0 = (int)y0f;
    const int x1 = x0 + 1,   y1 = y0 + 1;

    const bool xv0 = (unsigned)x0 < (unsigned)W_;
    const bool xv1 = (unsigned)x1 < (unsigned)W_;
    const bool yv0 = (unsigned)y0 < (unsigned)H_;
    const bool yv1 = (unsigned)y1 < (unsigned)H_;

    const int cx0 = min(max(x0, 0), W_ - 1);
    const int cx1 = min(max(x1, 0), W_ - 1);
    const int cy0 = min(max(y0, 0), H_ - 1);
    const int cy1 = min(max(y1, 0), H_ - 1);

    const float* r0 = img + (size_t)cy0 * W_ * C_;
    const float* r1 = img + (size_t)cy1 * W_ * C_;

    const float v00 = (yv0 && xv0) ? r0[cx0 * C_ + lane] : 0.f;
    const float v01 = (yv0 && xv1) ? r0[cx1 * C_ + lane] : 0.f;
    const float v10 = (yv1 && xv0) ? r1[cx0 * C_ + lane] : 0.f;
    const float v11 = (yv1 && xv1) ? r1[cx1 * C_ + lane] : 0.f;

    const float omwx = 1.f - wx, omwy = 1.f - wy;
    const float top = v00 * omwx + v01 * wx;   // same expression order as ref
    const float bot = v10 * omwx + v11 * wx;
    orow[xi * C_ + lane] = top * omwy + bot * wy;
  }
}

// ---------------------------------------------------------------------------
// Workspace layout (floats): [0, 32768)  chunk partials (32 x 1024)
//                            [32768, 33792) channel sums (32 x 32)
//                            [33792, 33920) per-batch params (32 x 4)
// ---------------------------------------------------------------------------
extern "C" void kernel_launch(void* const* d_in, const int* in_sizes, int n_in,
                              void* d_out, int out_size, void* d_ws, size_t ws_size,
                              hipStream_t stream) {
  const float* x    = (const float*)d_in[0];
  const float* Wm   = (const float*)d_in[1];
  const float* bias = (const float*)d_in[2];
  float* out = (float*)d_out;

  float* partial = (float*)d_ws;            // 32768 floats
  float* sums    = partial + CHUNKS * B_ * C_;  // 1024 floats
  float* params  = sums + B_ * C_;          // 128 floats

  LR_mean_partial<<<dim3(CHUNKS, B_), 256, 0, stream>>>(x, partial);
  LR_reduce_chunks<<<4, 256, 0, stream>>>(partial, sums);
  LR_angle<<<1, 32, 0, stream>>>(Wm, bias, sums, params);
  LR_rotate<<<dim3(B_, H_), 256, 0, stream>>>(x, params, out);
}